// ContourPostProcessor_76244259439040
// MI455X (gfx1250) — compile-verified
//
#include <hip/hip_runtime.h>
#include <hip/hip_bf16.h>

// Problem constants
#define BATCH 32
#define NQ    2000
#define NC    80
#define NCFLAT (NQ * NC)   // 160000 (divisible by 4)
#define TOPK  300
#define NP    360          // 720 floats = 180 float4 per contour row
#define NBINS 4096
#define CAND_CAP 2048
#define SORT_N   2048

// ---------- helpers ----------
__device__ __forceinline__ unsigned floatToKey(float f) {
    unsigned u = __float_as_uint(f);
    // monotone ascending mapping (bigger float -> bigger key)
    return u ^ ((u & 0x80000000u) ? 0xFFFFFFFFu : 0x80000000u);
}
__device__ __forceinline__ float keyToFloat(unsigned k) {
    unsigned u = (k & 0x80000000u) ? (k ^ 0x80000000u) : ~k;
    return __uint_as_float(u);
}

// ---------- pass 0: zero workspace ----------
__global__ void cp_zero_kernel(unsigned* hist, unsigned* cnt) {
    int i = blockIdx.x * blockDim.x + threadIdx.x;
    if (i < BATCH * NBINS) hist[i] = 0u;
    if (i < BATCH) cnt[i] = 0u;
}

// ---------- pass 1: per-batch histogram of key>>20 (LDS hist, then merge) ----------
__global__ void cp_hist_kernel(const float* __restrict__ logits,
                               unsigned* __restrict__ hist) {
    __shared__ unsigned lh[NBINS];
    const int b = blockIdx.y;
    for (int i = threadIdx.x; i < NBINS; i += blockDim.x) lh[i] = 0u;
    __syncthreads();
    const float4* base4 = (const float4*)(logits + (size_t)b * NCFLAT);
    const int n4 = NCFLAT / 4;
    const int stride = gridDim.x * blockDim.x;
    for (int i = blockIdx.x * blockDim.x + threadIdx.x; i < n4; i += stride) {
        float4 v = base4[i];
        atomicAdd(&lh[floatToKey(v.x) >> 20], 1u);
        atomicAdd(&lh[floatToKey(v.y) >> 20], 1u);
        atomicAdd(&lh[floatToKey(v.z) >> 20], 1u);
        atomicAdd(&lh[floatToKey(v.w) >> 20], 1u);
    }
    __syncthreads();
    unsigned* hb = hist + (size_t)b * NBINS;
    for (int i = threadIdx.x; i < NBINS; i += blockDim.x) {
        unsigned c = lh[i];
        if (c) atomicAdd(&hb[i], c);
    }
}

// ---------- pass 2: threshold bin per batch (block-parallel suffix scan) ----------
__global__ void __launch_bounds__(256)
cp_scan_kernel(const unsigned* __restrict__ hist,
               unsigned* __restrict__ thresh) {
    __shared__ unsigned part[256];
    const int b = blockIdx.x;
    const int t = threadIdx.x;
    const unsigned* hb = hist + (size_t)b * NBINS;
    // chunk t covers bins [top-15 .. top], top-down ordering (chunk 0 = highest bins)
    const int top = NBINS - 1 - t * 16;
    unsigned chunkCnt[16];
    unsigned s = 0;
    #pragma unroll
    for (int i = 0; i < 16; ++i) { chunkCnt[i] = hb[top - i]; s += chunkCnt[i]; }
    part[t] = s;
    __syncthreads();
    // inclusive scan over chunks (in top-down order)
    for (int off = 1; off < 256; off <<= 1) {
        unsigned add = (t >= off) ? part[t - off] : 0u;
        __syncthreads();
        part[t] += add;
        __syncthreads();
    }
    const unsigned incl = part[t];
    unsigned cum = incl - s;  // count strictly above my chunk
    if (cum < TOPK && incl >= TOPK) {
        // K-crossing is inside my chunk; walk 16 bins top-down
        #pragma unroll
        for (int i = 0; i < 16; ++i) {
            unsigned c = chunkCnt[i];
            if (cum + c >= TOPK) {
                thresh[2 * b]     = (unsigned)(top - i);  // inclusive threshold bin
                thresh[2 * b + 1] = cum;                  // count strictly above
                break;
            }
            cum += c;
        }
    }
}

// ---------- pass 3: collect candidates >= threshold bin ----------
__global__ void cp_collect_kernel(const float* __restrict__ logits,
                                  const unsigned* __restrict__ thresh,
                                  unsigned* __restrict__ cnt,
                                  unsigned long long* __restrict__ cand) {
    const int b = blockIdx.y;
    const unsigned T = thresh[2 * b];
    const float4* base4 = (const float4*)(logits + (size_t)b * NCFLAT);
    unsigned long long* cb = cand + (size_t)b * CAND_CAP;
    const int n4 = NCFLAT / 4;
    const int stride = gridDim.x * blockDim.x;
    for (int i = blockIdx.x * blockDim.x + threadIdx.x; i < n4; i += stride) {
        float4 v = base4[i];
        float e[4] = {v.x, v.y, v.z, v.w};
        #pragma unroll
        for (int c = 0; c < 4; ++c) {
            unsigned k = floatToKey(e[c]);
            if ((k >> 20) >= T) {
                unsigned pos = atomicAdd(&cnt[b], 1u);
                if (pos < CAND_CAP) {
                    // pack (key major, ~idx minor): descending sort gives
                    // score desc, index asc (jax.lax.top_k tie order)
                    unsigned idx = (unsigned)(4 * i + c);
                    cb[pos] = ((unsigned long long)k << 32) | (~idx);
                }
            }
        }
    }
}

// ---------- pass 4: per-batch bitonic sort (descending) + emit ----------
__global__ void __launch_bounds__(1024)
cp_sort_emit_kernel(const unsigned long long* __restrict__ cand,
                    const unsigned* __restrict__ cnt,
                    const float* __restrict__ boxes,       // [B,NQ,4] cxcywh
                    const float* __restrict__ sizes,       // [B,2] (W,H)
                    float* __restrict__ out_labels,        // [B,K]
                    float* __restrict__ out_boxes,         // [B,K,4]
                    float* __restrict__ out_scores,        // [B,K]
                    int* __restrict__ qidx_out) {          // [B,K]
    __shared__ unsigned long long s[SORT_N];
    const int b = blockIdx.x;
    unsigned count = cnt[b];
    if (count > CAND_CAP) count = CAND_CAP;
    const unsigned long long* cb = cand + (size_t)b * CAND_CAP;
    for (int i = threadIdx.x; i < SORT_N; i += blockDim.x)
        s[i] = (i < (int)count) ? cb[i] : 0ULL;
    __syncthreads();
    // bitonic sort, overall descending
    for (int k = 2; k <= SORT_N; k <<= 1) {
        for (int j = k >> 1; j > 0; j >>= 1) {
            for (int i = threadIdx.x; i < SORT_N; i += blockDim.x) {
                int ixj = i ^ j;
                if (ixj > i) {
                    unsigned long long a = s[i], c = s[ixj];
                    bool up = ((i & k) == 0);
                    if (up ? (a < c) : (a > c)) { s[i] = c; s[ixj] = a; }
                }
            }
            __syncthreads();
        }
    }
    const int r = threadIdx.x;
    if (r < TOPK) {
        unsigned long long e = s[r];
        unsigned key = (unsigned)(e >> 32);
        unsigned idx = (unsigned)(e & 0xFFFFFFFFu) ^ 0xFFFFFFFFu;
        if (idx >= (unsigned)NCFLAT) idx = 0u;  // defensive (padding never reached)
        float logit = keyToFloat(key);
        float score = 1.0f / (1.0f + expf(-logit));
        int lab = (int)(idx % (unsigned)NC) + 1;
        int q   = (int)(idx / (unsigned)NC);
        const size_t o = (size_t)b * TOPK + r;
        out_labels[o] = (float)lab;
        out_scores[o] = score;
        qidx_out[o]   = q;
        const float W = sizes[2 * b], H = sizes[2 * b + 1];
        const float* bx = boxes + ((size_t)b * NQ + q) * 4;
        float cx = bx[0], cy = bx[1], w = bx[2], h = bx[3];
        float* ob = out_boxes + o * 4;
        ob[0] = (cx - 0.5f * w) * W;
        ob[1] = (cy - 0.5f * h) * H;
        ob[2] = (cx + 0.5f * w) * W;
        ob[3] = (cy + 0.5f * h) * H;
    }
}

// ---------- pass 5: coords gather, one wave per row, pipelined async->LDS ----------
#define ROWS_PER_BLOCK 8   // 256 threads = 8 wave32, one contour row per wave
__global__ void __launch_bounds__(256)
cp_coords_kernel(const float* __restrict__ coords,   // [B,NQ,NP,2]
                 const int* __restrict__ qidx,       // [B,K]
                 const float* __restrict__ sizes,    // [B,2]
                 float* __restrict__ out) {          // [B,K,NP,2]
    __shared__ float4 stage[ROWS_PER_BLOCK * 180];
    const int wave = threadIdx.x >> 5;
    const int lane = threadIdx.x & 31;
    const int row = blockIdx.x * ROWS_PER_BLOCK + wave;   // grid covers exactly B*K
    const int b = row / TOPK;
    const int q = qidx[row];
    const float W = sizes[2 * b], H = sizes[2 * b + 1];
    const float4* src = (const float4*)(coords + ((size_t)b * NQ + q) * (NP * 2));
    float4* dst = (float4*)(out + (size_t)row * (NP * 2));
    float4* st = &stage[wave * 180];
    // issue all async global->LDS copies for this row (up to 6 in flight / lane)
    for (int i = lane; i < 180; i += 32) {
        unsigned ldsOff = (unsigned)(unsigned long long)(&st[i]);
        unsigned long long gaddr = (unsigned long long)(const void*)(src + i);
        asm volatile("global_load_async_to_lds_b128 %0, %1, off"
                     :: "v"(ldsOff), "v"(gaddr)
                     : "memory");
    }
    asm volatile("s_wait_asynccnt 0" ::: "memory");
    for (int i = lane; i < 180; i += 32) {
        float4 v = st[i];
        v.x *= W; v.y *= H; v.z *= W; v.w *= H;
        dst[i] = v;
    }
}

extern "C" void kernel_launch(void* const* d_in, const int* in_sizes, int n_in,
                              void* d_out, int out_size, void* d_ws, size_t ws_size,
                              hipStream_t stream) {
    const float* logits = (const float*)d_in[0];   // [B,NQ,NC]
    const float* coords = (const float*)d_in[1];   // [B,NQ,NP,2]
    const float* boxes  = (const float*)d_in[2];   // [B,NQ,4]
    const float* osize  = (const float*)d_in[3];   // [B,2]
    // d_in[4] (input_sizes) unused by the ratio_scale=False branch

    // Output layout: labels [B,K] | boxes [B,K,4] | coords [B,K,NP,2] | scores [B,K]
    float* out_labels = (float*)d_out;
    float* out_boxes  = out_labels + BATCH * TOPK;
    float* out_coords = out_boxes + (size_t)BATCH * TOPK * 4;
    float* out_scores = out_coords + (size_t)BATCH * TOPK * NP * 2;

    // Workspace layout (re-initialized every call; ~1.04 MB)
    unsigned* hist   = (unsigned*)d_ws;                        // 32*4096
    unsigned* thresh = hist + BATCH * NBINS;                   // 64
    unsigned* cnt    = thresh + 2 * BATCH;                     // 32
    unsigned long long* cand = (unsigned long long*)(cnt + BATCH); // 8B-aligned
    int* qidx = (int*)(cand + (size_t)BATCH * CAND_CAP);       // 32*300

    cp_zero_kernel<<<(BATCH * NBINS + 255) / 256, 256, 0, stream>>>(hist, cnt);
    cp_hist_kernel<<<dim3(64, BATCH), 256, 0, stream>>>(logits, hist);
    cp_scan_kernel<<<BATCH, 256, 0, stream>>>(hist, thresh);
    cp_collect_kernel<<<dim3(64, BATCH), 256, 0, stream>>>(logits, thresh, cnt, cand);
    cp_sort_emit_kernel<<<BATCH, 1024, 0, stream>>>(cand, cnt, boxes, osize,
                                                    out_labels, out_boxes,
                                                    out_scores, qidx);
    cp_coords_kernel<<<(BATCH * TOPK) / ROWS_PER_BLOCK, 256, 0, stream>>>(
        coords, qidx, osize, out_coords);
}